// GraphConv_p_2018634629393
// MI455X (gfx1250) — compile-verified
//
#include <hip/hip_runtime.h>
#include <hip/hip_bf16.h>

// Problem constants (match reference)
constexpr int BB   = 256;
constexpr int NN   = 512;
constexpr int DD   = 6;     // max neighbors per node
constexpr int FD   = 128;   // F_IN == F_OUT
constexpr int MAXD = 6;
constexpr int BN   = BB * NN;          // 131072 flat nodes
constexpr int ROWS_PER_BLOCK = 128;    // 8 waves x 16 rows
constexpr int MB   = (BN + ROWS_PER_BLOCK - 1) / ROWS_PER_BLOCK;  // 1024

// LDS strides (floats), padded to avoid 64-bank conflicts at row stride 512B.
// WS_STRIDE = 136 is produced by the TDM pad feature (128 dwords + 8 dword pad).
constexpr int AS_STRIDE = 132;
constexpr int WS_STRIDE = 136;
constexpr size_t LDS_BYTES =
    (size_t)(ROWS_PER_BLOCK * AS_STRIDE + FD * WS_STRIDE) * sizeof(float)
    + ROWS_PER_BLOCK * sizeof(int);

typedef __attribute__((ext_vector_type(2))) float v2f;
typedef __attribute__((ext_vector_type(8))) float v8f;
typedef __attribute__((ext_vector_type(4))) unsigned int u32x4;
typedef __attribute__((ext_vector_type(4))) int i32x4;
typedef __attribute__((ext_vector_type(8))) int i32x8;

// ---------------------------------------------------------------------------
// Kernel 1: per-node degree + compaction into per-degree buckets.
// Bucket fill order is atomics-nondeterministic but the final per-node output
// values are identical regardless of order.
// ---------------------------------------------------------------------------
__global__ void ngfp_bucket_kernel(const int* __restrict__ edges,
                                   int* __restrict__ cnt,
                                   int* __restrict__ bucket) {
  int i = blockIdx.x * blockDim.x + threadIdx.x;
  if (i >= BN) return;
  int deg = 0;
#pragma unroll
  for (int t = 0; t < DD; ++t) deg += (edges[i * DD + t] >= 0) ? 1 : 0;
  if (deg > MAXD - 1) deg = MAXD - 1;
  int pos = atomicAdd(&cnt[deg], 1);
  bucket[deg * BN + pos] = i;
}

// ---------------------------------------------------------------------------
// Kernel 2: per (degree d, tile of 128 bucket rows):
//   - wave 0 issues a TDM tensor_load_to_lds for the W[d] 128x128 f32 tile,
//     with D# padding producing LDS row stride 136 floats (bank-conflict pad)
//   - all waves gather+sum neighbor rows for 128 nodes into LDS (A tile),
//     fully overlapped with the TDM DMA
//   - 8 waves x (16x128) output via V_WMMA_F32_16X16X4_F32, K-loop of 32
//   - bias + relu + scatter stores to out[node]
// ---------------------------------------------------------------------------
__global__ __launch_bounds__(256)
void ngfp_gemm_kernel(const float* __restrict__ atoms,
                      const int*   __restrict__ edges,
                      const float* __restrict__ W,
                      const float* __restrict__ bias,
                      const int*   __restrict__ cnt,
                      const int*   __restrict__ bucket,
                      float* __restrict__ out) {
  const int d  = blockIdx.x / MB;
  const int mb = blockIdx.x % MB;
  const int count = cnt[d];
  const int rows_base = mb * ROWS_PER_BLOCK;
  if (rows_base >= count) return;   // block-uniform early exit

  extern __shared__ float smem[];
  float* As    = smem;                                   // [128][AS_STRIDE]
  float* Ws    = smem + ROWS_PER_BLOCK * AS_STRIDE;      // [128][WS_STRIDE]
  int*   nodes = (int*)(Ws + FD * WS_STRIDE);            // [128]

  const int tid  = threadIdx.x;
  const int lane = tid & 31;
  const int wv   = tid >> 5;        // wave id 0..7
  const int half = lane >> 4;       // 0 or 1 (lane half)
  const int l15  = lane & 15;

  // ---- wave 0: launch TDM DMA of W[d] (128x128 f32) into Ws ----
  // D# group0: count=1, lds_addr, global_addr, type=2
  // D# group1: data_size=4B, pad_enable, pad_interval=6 (128 dwords),
  //            pad_amount=7 (8 dwords) -> LDS row stride 136 floats,
  //            tensor_dim0=tensor_dim1=128, tile_dim0=tile_dim1=128,
  //            tensor_dim0_stride=128
  if (wv == 0) {
    unsigned long long ga =
        (unsigned long long)(const void*)(W + (size_t)d * FD * FD);
    unsigned int lds_off = (unsigned int)((const char*)Ws - (const char*)smem);
    u32x4 g0;
    g0[0] = 1u;                                          // count=1, user mode
    g0[1] = lds_off;                                     // lds_addr
    g0[2] = (unsigned int)(ga & 0xFFFFFFFFull);          // global_addr[31:0]
    g0[3] = (unsigned int)((ga >> 32) & 0x1FFFFFFull)    // global_addr[56:32]
            | (2u << 30);                                // type=2 (image)
    i32x8 g1;
    g1[0] = (int)((2u << 16)       // data_size = 4 bytes
                | (1u << 20)       // pad_enable
                | (6u << 22)       // pad_interval: 2^(6+1)=128 dwords
                | (7u << 25));     // pad_amount: 7+1 = 8 dwords
    g1[1] = (int)(128u << 16);     // tensor_dim0 = 128 (low 16 in [31:16])
    g1[2] = (int)(128u << 16);     // tensor_dim1 = 128 (low 16 in [31:16])
    g1[3] = (int)(128u << 16);     // tile_dim0 = 128  ([127:112])
    g1[4] = 128;                   // tile_dim1 = 128  ([143:128])
    g1[5] = 128;                   // tensor_dim0_stride = 128 ([191:160])
    g1[6] = 0;
    g1[7] = 0;
    i32x4 z4 = {0, 0, 0, 0};
#if __has_include(<hip/amd_detail/amd_gfx1250_TDM.h>)
    i32x8 z8 = {0, 0, 0, 0, 0, 0, 0, 0};
    __builtin_amdgcn_tensor_load_to_lds(g0, g1, z4, z4, z8, 0);
#else
    __builtin_amdgcn_tensor_load_to_lds(g0, g1, z4, z4, 0);
#endif
  }

  // ---- stage A tile (overlapped with TDM): gather + sum rows ----
  {
    const int* bkt = bucket + d * BN + rows_base;
    for (int j = wv; j < ROWS_PER_BLOCK; j += 8) {
      int row_global = rows_base + j;
      int node = (row_global < count) ? bkt[j] : 0;   // pad rows masked at store
      if (lane == 0) nodes[j] = node;
      int bat = node >> 9;                            // node / N (N = 512)
      const float* arow = atoms + (size_t)node * FD;  // flat [B,N,F]
      float4 a4 = *(const float4*)(arow + lane * 4);
      const int* erow = edges + (size_t)node * DD;
      for (int t = 0; t < d; ++t) {                   // bucket d => d valid slots
        int e = erow[t];
        if (e >= 0) {
          const float* nrow = atoms + ((size_t)(bat * NN + e)) * FD;
          float4 v = *(const float4*)(nrow + lane * 4);
          a4.x += v.x; a4.y += v.y; a4.z += v.z; a4.w += v.w;
        }
      }
      *(float4*)(As + j * AS_STRIDE + lane * 4) = a4;
    }
  }

  if (wv == 0) __builtin_amdgcn_s_wait_tensorcnt(0);  // W[d] resident in LDS
  __syncthreads();

  // ---- WMMA GEMM: this wave computes rows [wv*16, wv*16+16) x all 128 cols ----
  // A frag (16x4 f32): lane m=l15 holds K = k0+2*half, k0+2*half+1
  // B frag (4x16 f32): lane n=l15 holds same K pair
  const float* Arow = As + (wv * 16 + l15) * AS_STRIDE;

  v8f acc[8] = {};
#pragma unroll 4
  for (int ks = 0; ks < FD / 4; ++ks) {
    int k0 = ks * 4 + half * 2;
    v2f a;
    a.x = Arow[k0];
    a.y = Arow[k0 + 1];
#pragma unroll
    for (int ct = 0; ct < 8; ++ct) {
      int col = ct * 16 + l15;
      v2f b;
      b.x = Ws[k0 * WS_STRIDE + col];
      b.y = Ws[(k0 + 1) * WS_STRIDE + col];
      // 8 args: (neg_a, A, neg_b, B, c_mod, C, reuse_a, reuse_b)
      acc[ct] = __builtin_amdgcn_wmma_f32_16x16x4_f32(
          false, a, false, b, (short)0, acc[ct], false, false);
    }
  }

  // ---- epilogue: bias + relu + scatter (C/D layout: M = r + 8*half, N = l15) ----
  const float* bd = bias + (size_t)d * FD;
#pragma unroll
  for (int ct = 0; ct < 8; ++ct) {
    int col = ct * 16 + l15;
    float bb = bd[col];
#pragma unroll
    for (int r = 0; r < 8; ++r) {
      int m = r + half * 8;
      int row_global = rows_base + wv * 16 + m;
      if (row_global < count) {
        int node = nodes[wv * 16 + m];
        float v = acc[ct][r] + bb;
        out[(size_t)node * FD + col] = v > 0.0f ? v : 0.0f;
      }
    }
  }
}

// ---------------------------------------------------------------------------
// Host launcher
// d_in[0] = atoms  [B,N,F_IN] f32      d_in[1] = edges [B,N,D] int32
// d_in[2] = W      [6,128,128] f32     d_in[3] = b     [6,128]  f32
// d_out   = out    [B,N,F_OUT] f32
// d_ws    : [0,24)  cnt[6] ; [256, 256 + 6*BN*4) buckets  (~3.1 MB total)
// ---------------------------------------------------------------------------
extern "C" void kernel_launch(void* const* d_in, const int* in_sizes, int n_in,
                              void* d_out, int out_size, void* d_ws, size_t ws_size,
                              hipStream_t stream) {
  const float* atoms = (const float*)d_in[0];
  const int*   edges = (const int*)d_in[1];   // jax default x64-off -> int32
  const float* W     = (const float*)d_in[2];
  const float* bias  = (const float*)d_in[3];
  float*       out   = (float*)d_out;

  int* cnt    = (int*)d_ws;
  int* bucket = (int*)((char*)d_ws + 256);

  hipMemsetAsync(d_ws, 0, 256, stream);   // zero counters (graph-capturable)

  ngfp_bucket_kernel<<<(BN + 255) / 256, 256, 0, stream>>>(edges, cnt, bucket);

  ngfp_gemm_kernel<<<MAXD * MB, 256, LDS_BYTES, stream>>>(
      atoms, edges, W, bias, cnt, bucket, out);
}